// GCNLinear_64390149702456
// MI455X (gfx1250) — compile-verified
//
#include <hip/hip_runtime.h>
#include <hip/hip_bf16.h>

typedef __attribute__((ext_vector_type(2))) float v2f;
typedef __attribute__((ext_vector_type(8))) float v8f;

#define D_FEAT 64

// ---------------------------------------------------------------------------
// Kernel 1: zero the accumulation buffer h[N,64] (workspace is poisoned 0xAA).
// ---------------------------------------------------------------------------
__global__ void gcn_zero_kernel(float4* __restrict__ h4, int n4) {
    int i = blockIdx.x * blockDim.x + threadIdx.x;
    if (i < n4) {
        h4[i] = make_float4(0.f, 0.f, 0.f, 0.f);
    }
}

// ---------------------------------------------------------------------------
// Kernel 2: edge-parallel scatter-add. One wave32 per edge; each lane moves
// a float2 (32 lanes x 2 = 64 features). Gather is a coalesced b64 load from
// the L2-resident feature table; accumulation is global_atomic_add_f32.
// ---------------------------------------------------------------------------
__global__ void gcn_scatter_kernel(const float* __restrict__ feat,
                                   const int* __restrict__ src,
                                   const int* __restrict__ dst,
                                   float* __restrict__ h,
                                   int n_edges) {
    const int wave = (blockIdx.x * blockDim.x + threadIdx.x) >> 5;
    const int lane = threadIdx.x & 31;
    if (wave >= n_edges) return;            // uniform per wave

    const int s = src[wave];
    const int d = dst[wave];

    const v2f v = *(const v2f*)(feat + (long)s * D_FEAT + lane * 2);
    float* p = h + (long)d * D_FEAT + lane * 2;
    atomicAdd(p,     v.x);
    atomicAdd(p + 1, v.y);
}

// ---------------------------------------------------------------------------
// Kernel 3: out = h @ W^T + b using full-precision V_WMMA_F32_16X16X4_F32.
//
// One wave per 16-row M-tile. N=64 -> 4 accumulator tiles (v8f each).
// K=64 -> 16 WMMA steps of K=4.
//
// 32-bit A(16x4) fragment layout: lanes 0-15 hold K={k0,k0+1} of row M=lane,
// lanes 16-31 hold K={k0+2,k0+3} of row M=lane-16  => contiguous v2f load
// from h[row][kk]. B = W^T (B[k][n] = W[n][k]) mirrors this: contiguous v2f
// load from W[n][kk]. C/D(16x16): VGPR r holds (M=r, N=lane) for lanes 0-15
// and (M=r+8, N=lane-16) for lanes 16-31.
// ---------------------------------------------------------------------------
__global__ void gcn_gemm_wmma_kernel(const float* __restrict__ h,
                                     const float* __restrict__ W,
                                     const float* __restrict__ bias,
                                     float* __restrict__ out,
                                     int n_tiles) {
    const int lane   = threadIdx.x & 31;
    const int waveIb = threadIdx.x >> 5;
    const int tile   = blockIdx.x * (blockDim.x >> 5) + waveIb;
    if (tile >= n_tiles) return;            // uniform per wave; EXEC stays all-1s

    const int m0      = tile * 16;
    const int l15     = lane & 15;
    const int halfSel = (lane < 16) ? 0 : 2;

    v8f acc0 = {}, acc1 = {}, acc2 = {}, acc3 = {};

    const float* arow = h + (long)(m0 + l15) * D_FEAT;
    const float* wr0  = W + (long)( 0 + l15) * D_FEAT;
    const float* wr1  = W + (long)(16 + l15) * D_FEAT;
    const float* wr2  = W + (long)(32 + l15) * D_FEAT;
    const float* wr3  = W + (long)(48 + l15) * D_FEAT;

#pragma unroll
    for (int s = 0; s < 16; ++s) {
        const int kk = s * 4 + halfSel;
        const v2f a  = *(const v2f*)(arow + kk);
        const v2f b0 = *(const v2f*)(wr0 + kk);
        const v2f b1 = *(const v2f*)(wr1 + kk);
        const v2f b2 = *(const v2f*)(wr2 + kk);
        const v2f b3 = *(const v2f*)(wr3 + kk);
        // 8 args: (neg_a, A, neg_b, B, c_mod, C, reuse_a, reuse_b)
        acc0 = __builtin_amdgcn_wmma_f32_16x16x4_f32(false, a, false, b0,
                                                     (short)0, acc0, false, false);
        acc1 = __builtin_amdgcn_wmma_f32_16x16x4_f32(false, a, false, b1,
                                                     (short)0, acc1, false, false);
        acc2 = __builtin_amdgcn_wmma_f32_16x16x4_f32(false, a, false, b2,
                                                     (short)0, acc2, false, false);
        acc3 = __builtin_amdgcn_wmma_f32_16x16x4_f32(false, a, false, b3,
                                                     (short)0, acc3, false, false);
    }

    // Bias + store. D-tile t covers columns [16t, 16t+16).
    const int rowAdd = (lane < 16) ? 0 : 8;
    const float bb0 = bias[ 0 + l15];
    const float bb1 = bias[16 + l15];
    const float bb2 = bias[32 + l15];
    const float bb3 = bias[48 + l15];

#pragma unroll
    for (int r = 0; r < 8; ++r) {
        float* orow = out + (long)(m0 + r + rowAdd) * D_FEAT;
        orow[ 0 + l15] = acc0[r] + bb0;
        orow[16 + l15] = acc1[r] + bb1;
        orow[32 + l15] = acc2[r] + bb2;
        orow[48 + l15] = acc3[r] + bb3;
    }
}

// ---------------------------------------------------------------------------
// Launch: zero h -> scatter edges -> WMMA GEMM + bias. All on `stream`.
// ---------------------------------------------------------------------------
extern "C" void kernel_launch(void* const* d_in, const int* in_sizes, int n_in,
                              void* d_out, int out_size, void* d_ws, size_t ws_size,
                              hipStream_t stream) {
    const float* feature = (const float*)d_in[0];   // [N, 64] f32
    const int*   src     = (const int*)d_in[1];     // [E] int
    const int*   dst     = (const int*)d_in[2];     // [E] int
    const float* W       = (const float*)d_in[3];   // [64, 64] f32 (out, in)
    const float* bias    = (const float*)d_in[4];   // [64] f32

    float* out = (float*)d_out;                     // [N, 64] f32
    float* h   = (float*)d_ws;                      // [N, 64] f32 scratch

    const int n_nodes = in_sizes[0] / D_FEAT;
    const int n_edges = in_sizes[1];

    // 1) h = 0
    const int n4 = n_nodes * D_FEAT / 4;
    gcn_zero_kernel<<<(n4 + 255) / 256, 256, 0, stream>>>((float4*)h, n4);

    // 2) h[dst] += feature[src]   (one wave per edge, 8 edges per 256-thread block)
    const long total_threads = (long)n_edges * 32;
    const int  scat_blocks   = (int)((total_threads + 255) / 256);
    gcn_scatter_kernel<<<scat_blocks, 256, 0, stream>>>(feature, src, dst, h, n_edges);

    // 3) out = h @ W^T + b   (one wave per 16-row tile, 4 waves per block)
    const int n_tiles = (n_nodes + 15) / 16;        // 50000/16 = 3125 exact
    gcn_gemm_wmma_kernel<<<(n_tiles + 3) / 4, 128, 0, stream>>>(h, W, bias, out, n_tiles);
}